// FreeEnergyA_34686155883201
// MI455X (gfx1250) — compile-verified
//
#include <hip/hip_runtime.h>
#include <math.h>

// FreeEnergyA for MI455X (gfx1250, wave32).
// One workgroup per sample: trajectory scan -> LDS, Gram via v_wmma_f32_16x16x4_f32,
// branch-free erf/log likelihood on WMMA fragments, block reduce, global_atomic_add_f32.

#define NN 128      // N_NODES
#define DD 127      // N_NODES-1 (trajectory dims; col 127 is zero pad -> K=128)
#define STRIDE 132  // LDS row stride (dwords): 132%64==4 -> conflict-free, 528B row = 8B aligned
#define LMAX 0.06f
#define EPSP 1e-7f

typedef __attribute__((ext_vector_type(2))) float v2f;
typedef __attribute__((ext_vector_type(8))) float v8f;

// ---- raw hardware transcendentals (single TRANS ops, no libm branches) ----
__device__ __forceinline__ float fexp(float x) {           // e^x
  return __builtin_amdgcn_exp2f(x * 1.44269504088896341f);
}
__device__ __forceinline__ float flog(float x) {           // ln x
  return __builtin_amdgcn_logf(x) * 0.69314718055994531f;
}
__device__ __forceinline__ float fsqrt(float x)  { return __builtin_amdgcn_sqrtf(x); }
__device__ __forceinline__ float frsqrt(float x) { return __builtin_amdgcn_rsqf(x); }
__device__ __forceinline__ float frcp(float x)   { return __builtin_amdgcn_rcpf(x); }

__device__ __forceinline__ float softplus_f(float x) {
  return (x > 20.f) ? x : flog(1.f + fexp(x));
}

// erf for x >= 0, branch-free (Abramowitz-Stegun 7.1.26, |err| < 1.5e-7)
__device__ __forceinline__ float erf_pos(float x) {
  float t = frcp(fmaf(0.3275911f, x, 1.f));
  float p = fmaf(t, 1.061405429f, -1.453152027f);
  p = fmaf(t, p, 1.421413741f);
  p = fmaf(t, p, -0.284496736f);
  p = fmaf(t, p, 0.254829592f);
  return 1.f - t * p * fexp(-x * x);
}
// Phi(z) for z >= 0
__device__ __forceinline__ float ncdf_pos(float z) {
  return 0.5f * (1.f + erf_pos(z * 0.7071067811865475f));
}

// one barrier-round reduction of a pair of values across 256 threads
__device__ __forceinline__ void block_reduce2(float& x, float& y, float* red,
                                              int lane, int wid) {
#pragma unroll
  for (int off = 16; off; off >>= 1) {
    x += __shfl_xor(x, off, 32);
    y += __shfl_xor(y, off, 32);
  }
  if (lane == 0) { red[2 * wid] = x; red[2 * wid + 1] = y; }
  __syncthreads();
  float sx = 0.f, sy = 0.f;
#pragma unroll
  for (int w = 0; w < 8; ++w) { sx += red[2 * w]; sy += red[2 * w + 1]; }
  x = sx; y = sy;
  __syncthreads();  // red[] reused next round
}

__global__ void fe_zero_kernel(float* out) { out[0] = 0.f; }

__global__ __launch_bounds__(256) void free_energy_kernel(
    const float* __restrict__ d_raw,   // [S, 127]
    const float* __restrict__ c_raw,   // [S, 126]
    const float* __restrict__ a_raw,   // [S, 126, 127]
    const float* __restrict__ l_raw,   // [S, 1]
    const int*   __restrict__ all_tr,  // [128, 128]
    const int*   __restrict__ cor_tr,  // [128, 128]
    float* __restrict__ out, int S)
{
  __shared__ float xs[NN * STRIDE];  // trajectory, row-major, K padded to 128
  __shared__ float sq[NN];           // per-node squared norms
  __shared__ float red[16];
  __shared__ float lam_sh;

  const int s    = blockIdx.x;
  const int tid  = threadIdx.x;
  const int lane = tid & 31;
  const int wid  = tid >> 5;

  // warm L2/WGP$ for the shared trial tables (global_prefetch_b8)
  __builtin_prefetch(all_tr + tid * 64, 0, 0);
  __builtin_prefetch(cor_tr + tid * 64, 0, 0);
  if (tid == 0) lam_sh = LMAX * 0.5f * (1.f + erff(l_raw[s] * 0.7071067811865475f));

  // ---------------- Phase 1: trajectory recurrence (serial over nodes) -------
  const bool act = (tid < DD);   // dim owner; tid==127 is the zero pad column
  float v = 0.f, x = 0.f;
  if (tid < NN) {
    float d0 = softplus_f(d_raw[(size_t)s * DD + 0]);
    v = (tid == 0) ? 1.f : 0.f;          // e0
    x = d0 * v;                          // x1 = d0 * e0
    xs[0 * STRIDE + tid] = 0.f;          // node 0 at origin
    xs[1 * STRIDE + tid] = x;            // node 1
  }
  __syncthreads();

  const size_t abase = (size_t)s * (NN - 2) * DD;
  // software-pipelined a_raw stream: fetch i+1 before the reductions of step i
  float a = act ? a_raw[abase + tid] : 0.f;
  for (int i = 0; i < NN - 2; ++i) {
    float a_nxt = (act && i + 1 < NN - 2) ? a_raw[abase + (size_t)(i + 1) * DD + tid] : 0.f;
    // v_prev is unit, so ||a - (v.a)v||^2 == ||a||^2 - (v.a)^2 -> ONE reduction round
    float va = v * a, saa = a * a;
    block_reduce2(va, saa, red, lane, wid);
    float n2  = fmaxf(saa - va * va, 1e-30f);
    float inv = frsqrt(n2);
    float w   = a - va * v;
    float ci  = c_raw[(size_t)s * (NN - 2) + i];
    float di  = softplus_f(d_raw[(size_t)s * DD + i + 1]);
    float sc, cc;
    __sincosf(ci, &sc, &cc);
    float vn = cc * v + sc * (w * inv);
    if (!act) vn = 0.f;
    v = vn;
    x = x + di * v;
    if (tid < NN) xs[(i + 2) * STRIDE + tid] = x;
    a = a_nxt;
  }
  __syncthreads();

  // per-node squared norms
  if (tid < NN) {
    float acc = 0.f;
#pragma unroll 4
    for (int d = 0; d < NN; ++d) {
      float t = xs[tid * STRIDE + d];
      acc += t * t;
    }
    sq[tid] = acc;
  }
  __syncthreads();

  // ---------------- Phase 2: Gram via FP32 WMMA + likelihood -----------------
  const float lam = lam_sh;
  const float one_m2l = 1.f - 2.f * lam;
  float ll_acc = 0.f;

  const int ti = wid;            // wave w owns output tile-row w
  const int hl = lane >> 4;      // half-wave select (K pair)
  const int li = lane & 15;

  // A fragments (16x4 f32) for the whole K range, cached in registers:
  // lane holds X[ti*16+li][4k+2*hl .. +1] -> reused across all 8 tj tiles
  const float* pa = &xs[(ti * 16 + li) * STRIDE + 2 * hl];
  v2f areg[NN / 4];
#pragma unroll
  for (int k = 0; k < NN / 4; ++k) areg[k] = *(const v2f*)(pa + 4 * k);

  for (int tj = 0; tj < 8; ++tj) {
    const float* pb = &xs[(tj * 16 + li) * STRIDE + 2 * hl];  // B = X^T, same addressing
    v8f c8 = {};
#pragma unroll
    for (int k = 0; k < NN / 4; ++k) {
      v2f bf = *(const v2f*)(pb + 4 * k);   // ds_load_b64
      c8 = __builtin_amdgcn_wmma_f32_16x16x4_f32(
          false, areg[k], false, bf, (short)0, c8, false, false);
    }
    const int j = tj * 16 + li;
    const float sqj = sq[j];
#pragma unroll
    for (int r = 0; r < 8; ++r) {
      const int i = ti * 16 + r + 8 * hl;   // C/D layout: VGPR r -> M = r (+8 hi half)
      float g    = c8[r];
      float d2   = sq[i] + sqj - 2.f * g;
      float dist = fsqrt(fmaxf(d2, 0.f));   // v_sqrt_f32
      float pA = ncdf_pos(dist * 0.7071067811865475f);  // Phi(d/sqrt2)
      float pB = ncdf_pos(dist * 0.5f);                 // Phi(d/2)
      float paxb = pA * pB + (1.f - pA) * (1.f - pB);
      float pr = one_m2l * paxb + lam;
      pr = fminf(fmaxf(pr, EPSP), 1.f - EPSP);
      const int idx = i * NN + j;
      float ncor = (float)cor_tr[idx];
      float nall = (float)all_tr[idx];
      ll_acc += ncor * flog(pr) + (nall - ncor) * flog(1.f - pr);
    }
  }

  // ---------------- Final reduction -> mean over S ---------------------------
#pragma unroll
  for (int off = 16; off; off >>= 1) ll_acc += __shfl_xor(ll_acc, off, 32);
  if (lane == 0) red[wid] = ll_acc;
  __syncthreads();
  if (tid == 0) {
    float t = 0.f;
#pragma unroll
    for (int w8 = 0; w8 < 8; ++w8) t += red[w8];
    atomicAdd(out, t * (1.f / (float)S));
  }
}

extern "C" void kernel_launch(void* const* d_in, const int* in_sizes, int n_in,
                              void* d_out, int out_size, void* d_ws, size_t ws_size,
                              hipStream_t stream) {
  const float* d_raw = (const float*)d_in[0];
  const float* c_raw = (const float*)d_in[1];
  const float* a_raw = (const float*)d_in[2];
  const float* l_raw = (const float*)d_in[3];
  const int* all_tr  = (const int*)d_in[4];
  const int* cor_tr  = (const int*)d_in[5];
  float* out = (float*)d_out;

  const int S = in_sizes[3];  // l_raw is [S,1]

  fe_zero_kernel<<<1, 1, 0, stream>>>(out);
  free_energy_kernel<<<S, 256, 0, stream>>>(d_raw, c_raw, a_raw, l_raw,
                                            all_tr, cor_tr, out, S);
}